// LogHausdorffDTLoss_10514079941329
// MI455X (gfx1250) — compile-verified
//
#include <hip/hip_runtime.h>
#include <math.h>

// Problem constants (B=8, C=4, H=W=256)
#define HW      65536
#define NB      8
#define NCC     3              // channels 1..3 kept
#define NSLICE  24             // NB*NCC
#define NTOT    1572864        // NB*NCC*HW
#define GINF    512            // reference INF = H+W

typedef float v2f __attribute__((ext_vector_type(2)));
typedef float v8f __attribute__((ext_vector_type(8)));

// ---------------- workspace layout (bytes) ----------------
// [0,        6291456)  p       : 24*65536 f32   (softmax ch1..3)
// [6291456,  6815744)  labels  : 8*65536  u8    (argmax preds_T)
// [6815744,  6815936)  counts  : 48 u32  (0..23 pred mask popcount, 24..47 tgt)
// [6816000,  6816008)  acc     : 1 double
// [6816256, 19399168)  gbuf    : 48*65536 u32  (packed u16 g_mask | g_notmask<<16)
// total ~18.5 MB

__global__ void k_init(unsigned* __restrict__ counts, double* __restrict__ acc) {
  int t = threadIdx.x;
  if (t < 48) counts[t] = 0u;
  if (t == 48) *acc = 0.0;
}

// softmax over 4 channels, argmax labels, per-slice mask popcounts
__global__ void k_prep(const float* __restrict__ S, const float* __restrict__ T,
                       float* __restrict__ p, unsigned char* __restrict__ lab,
                       unsigned* __restrict__ counts) {
  __shared__ unsigned cnt[6];
  int t = threadIdx.x;
  if (t < 6) cnt[t] = 0u;
  __syncthreads();
  int idx = blockIdx.x * blockDim.x + t;      // 0 .. 524287, one (b,hw) pixel
  int b  = idx >> 16;                          // block spans a single b
  int hw = idx & 0xFFFF;

  const float* Sb = S + (size_t)b * (4 * HW) + hw;
  float s0 = Sb[0], s1 = Sb[HW], s2 = Sb[2 * HW], s3 = Sb[3 * HW];
  float m  = fmaxf(fmaxf(s0, s1), fmaxf(s2, s3));
  float e0 = expf(s0 - m), e1 = expf(s1 - m), e2 = expf(s2 - m), e3 = expf(s3 - m);
  float inv = 1.0f / (e0 + e1 + e2 + e3);
  float p1 = e1 * inv, p2 = e2 * inv, p3 = e3 * inv;
  p[(size_t)(b * 3 + 0) * HW + hw] = p1;
  p[(size_t)(b * 3 + 1) * HW + hw] = p2;
  p[(size_t)(b * 3 + 2) * HW + hw] = p3;
  if (p1 > 0.5f) atomicAdd(&cnt[0], 1u);
  if (p2 > 0.5f) atomicAdd(&cnt[1], 1u);
  if (p3 > 0.5f) atomicAdd(&cnt[2], 1u);

  const float* Tb = T + (size_t)b * (4 * HW) + hw;
  float t0 = Tb[0], t1 = Tb[HW], t2 = Tb[2 * HW], t3 = Tb[3 * HW];
  int L = 0; float bv = t0;                    // first-max tie break == jnp.argmax
  if (t1 > bv) { bv = t1; L = 1; }
  if (t2 > bv) { bv = t2; L = 2; }
  if (t3 > bv) { bv = t3; L = 3; }
  lab[idx] = (unsigned char)L;
  if (L >= 1) atomicAdd(&cnt[3 + (L - 1)], 1u);

  __syncthreads();
  if (t < 3)      atomicAdd(&counts[b * 3 + t], cnt[t]);
  else if (t < 6) atomicAdd(&counts[24 + b * 3 + (t - 3)], cnt[t]);
}

// vertical 1D distance (down + up scans) for mask and ~mask simultaneously.
// One block per slice (48), one thread per column. No cross-thread deps.
__global__ void k_vert(const float* __restrict__ p, const unsigned char* __restrict__ lab,
                       unsigned* __restrict__ g) {
  int s = blockIdx.x;                     // 0..47 (0..23 pred, 24..47 tgt)
  int w = threadIdx.x;                    // column
  bool isT = (s >= 24);
  int bc = isT ? s - 24 : s;
  int b = bc / 3, cc = bc % 3;
  unsigned* gs = g + (size_t)s * HW;

  int pm = GINF, pn = GINF;
  for (int h = 0; h < 256; ++h) {
    int off = h * 256 + w;
    bool m = isT ? (lab[b * HW + off] == (unsigned char)(cc + 1))
                 : (p[(size_t)bc * HW + off] > 0.5f);
    int fm = min(m ? GINF : 0, pm + 1);   // mask:  distance to nearest False
    int fn = min(m ? 0 : GINF, pn + 1);   // ~mask: distance to nearest True
    gs[off] = (unsigned)fm | ((unsigned)fn << 16);
    pm = fm; pn = fn;
  }
  pm = GINF; pn = GINF;
  for (int h = 255; h >= 0; --h) {
    int off = h * 256 + w;
    unsigned pk = gs[off];                // same-thread same-addr: ordered
    int gm = min((int)(pk & 0xFFFFu), pm + 1);
    int gn = min((int)(pk >> 16),     pn + 1);
    gs[off] = (unsigned)gm | ((unsigned)gn << 16);
    pm = gm; pn = gn;
  }
}

// per-row exact min-plus for 4 variants + fused loss reduction.
// grid (256 rows, 24 bc), 256 threads (one per output column j).
__global__ void k_row(const unsigned* __restrict__ g, const float* __restrict__ p,
                      const unsigned char* __restrict__ lab,
                      const unsigned* __restrict__ counts, double* __restrict__ acc) {
  __shared__ int   spm[256], spn[256], stm[256], stn[256];
  __shared__ float red[256];
  int h  = blockIdx.x;
  int bc = blockIdx.y;
  int b = bc / 3, cc = bc % 3;
  int j = threadIdx.x;

  {
    unsigned gp = g[(size_t)bc * HW + h * 256 + j];
    unsigned gt = g[(size_t)(24 + bc) * HW + h * 256 + j];
    int a = (int)(gp & 0xFFFFu), c = (int)(gp >> 16);
    spm[j] = a * a; spn[j] = c * c;
    a = (int)(gt & 0xFFFFu); c = (int)(gt >> 16);
    stm[j] = a * a; stn[j] = c * c;
  }
  __syncthreads();

  int mpm = 1 << 30, mpn = 1 << 30, mtm = 1 << 30, mtn = 1 << 30;
  for (int k = 0; k < 256; ++k) {        // tropical (min,+) inner product
    int t  = j - k;
    int d2 = t * t;
    mpm = min(mpm, spm[k] + d2);
    mpn = min(mpn, spn[k] + d2);
    mtm = min(mtm, stm[k] + d2);
    mtn = min(mtn, stn[k] + d2);
  }

  unsigned cp = counts[bc], ct = counts[24 + bc];
  bool vp = (cp > 0u) && (cp < (unsigned)HW);  // mask.any() && !mask.all()
  bool vt = (ct > 0u) && (ct < (unsigned)HW);
  int pdt2 = vp ? (mpm + mpn) : 0;             // field^2 = d2(mask)+d2(~mask)
  int tdt2 = vt ? (mtm + mtn) : 0;

  float pv = p[(size_t)bc * HW + h * 256 + j];
  float tv = (lab[b * HW + h * 256 + j] == (unsigned char)(cc + 1)) ? 1.0f : 0.0f;
  float diff = pv - tv;
  red[j] = diff * diff * (float)(pdt2 + tdt2);
  __syncthreads();

  for (int sft = 128; sft >= 32; sft >>= 1) {
    if (j < sft) red[j] += red[j + sft];
    __syncthreads();
  }
  if (j < 32) {                // one full wave32, EXEC all ones
    // WMMA cross-lane reduction: A(16x4) holds the 32 partials (K=0 lanes 0-15,
    // K=2 lanes 16-31; K=1/3 zero). B = ones(4x16) -> D[M][j] = p[M] + p[M+16].
    v2f A;  A[0] = red[j]; A[1] = 0.0f;
    v2f Bm; Bm[0] = 1.0f;  Bm[1] = 1.0f;
    v8f C = {};
    v8f D = __builtin_amdgcn_wmma_f32_16x16x4_f32(false, A, false, Bm,
                                                  (short)0, C, false, false);
    float ssum = D[0] + D[1] + D[2] + D[3] + D[4] + D[5] + D[6] + D[7];
    // lane 0 holds sum over M=0..7 of column 0; lane 16 holds M=8..15.
    if (j == 0 || j == 16) atomicAdd(acc, (double)ssum);
  }
}

__global__ void k_fin(const double* __restrict__ acc, float* __restrict__ out) {
  double loss = *acc / (double)NTOT;
  out[0] = (float)log(loss + 1.0);
}

extern "C" void kernel_launch(void* const* d_in, const int* in_sizes, int n_in,
                              void* d_out, int out_size, void* d_ws, size_t ws_size,
                              hipStream_t stream) {
  const float* S = (const float*)d_in[0];   // preds_S (8,4,256,256)
  const float* T = (const float*)d_in[1];   // preds_T (8,4,256,256)
  // d_in[2] (target) is unused by the reference loss.
  (void)in_sizes; (void)n_in; (void)out_size; (void)ws_size;

  char* ws = (char*)d_ws;
  float*         p      = (float*)(ws);
  unsigned char* lab    = (unsigned char*)(ws + 6291456);
  unsigned*      counts = (unsigned*)(ws + 6815744);
  double*        acc    = (double*)(ws + 6816000);
  unsigned*      g      = (unsigned*)(ws + 6816256);
  float*         out    = (float*)d_out;

  k_init<<<1, 64, 0, stream>>>(counts, acc);
  k_prep<<<2048, 256, 0, stream>>>(S, T, p, lab, counts);
  k_vert<<<48, 256, 0, stream>>>(p, lab, g);
  dim3 gr(256, 24);
  k_row<<<gr, 256, 0, stream>>>(g, p, lab, counts, acc);
  k_fin<<<1, 1, 0, stream>>>(acc, out);
}